// VectorBlock_46325517254761
// MI455X (gfx1250) — compile-verified
//
#include <hip/hip_runtime.h>
#include <hip/hip_bf16.h>

typedef __attribute__((ext_vector_type(2))) float v2f;
typedef __attribute__((ext_vector_type(8))) float v8f;

#define PSTRIDE 160  // dwords per k-row-pair in LDS; 160 % 64 == 32 -> conflict-free halves

// ---------------------------------------------------------------------------
// Generic tiled GEMM, fp32, using V_WMMA_F32_16X16X4_F32.
//   Cpart[z][2048 x Ncols] = A_slice[2048 x Kslice] * B_slice[Kslice x Ncols]
// A row-major (lda); B row-major (ldb) OR, if B_FROM_W, synthesized from
// W[cin, cout, T] as B[i, t*cout + j] = W[i*cout*T + j*T + t].
// Block: 256 threads = 8 waves. Block tile: 128 (M) x (NACC*16) (N, within a
// 64-wide LDS tile). K chunk: 64 staged in LDS with k-pair interleaving:
//   element (k-row r, col c) at Bs[(r>>1)*PSTRIDE + 2*c + (r&1)]
// so each WMMA B fragment (rows {brow, brow+1}, one col) is a single b64 load.
// ---------------------------------------------------------------------------
template <bool B_FROM_W, int NACC>
__global__ __launch_bounds__(256) void wmma_gemm_kernel(
    const float* __restrict__ A, int lda, int Ktotal, int Kslice,
    const float* __restrict__ Bsrc, int ldb, int T_, int coutw,
    float* __restrict__ C, int Ncols) {
  __shared__ float Bs[32 * PSTRIDE];  // 20 KB

  const int tid  = threadIdx.x;
  const int wave = tid >> 5;
  const int lane = tid & 31;
  const int lr   = lane & 15;   // lane-in-half
  const int lh   = lane >> 4;   // half id (0/1)

  const int m0    = blockIdx.x * 128 + wave * 16;  // 16 rows per wave
  const int nbase = blockIdx.y * 64;

  const int ks = blockIdx.z * Kslice;
  const int ke = (ks + Kslice < Ktotal) ? (ks + Kslice) : Ktotal;

  v8f acc[NACC];
#pragma unroll
  for (int nj = 0; nj < NACC; ++nj) acc[nj] = (v8f){};

  const float* arow = A + (size_t)(m0 + lr) * lda + 2 * lh;

  for (int kchunk = ks; kchunk < ke; kchunk += 64) {
    const int kc = (ke - kchunk) < 64 ? (ke - kchunk) : 64;  // 32 or 64 here

    // Cooperative LDS fill of B tile [kc x 64] (zero-padded past Ncols),
    // interleaved by k-row pair.
    for (int idx = tid; idx < kc * 64; idx += 256) {
      const int r  = idx >> 6;
      const int c  = idx & 63;
      const int gc = nbase + c;
      float v = 0.0f;
      if (gc < Ncols) {
        if (B_FROM_W) {
          const int t = gc / coutw;
          const int j = gc - t * coutw;
          v = Bsrc[(size_t)(kchunk + r) * coutw * T_ + j * T_ + t];
        } else {
          v = Bsrc[(size_t)(kchunk + r) * ldb + gc];
        }
      }
      Bs[(r >> 1) * PSTRIDE + 2 * c + (r & 1)] = v;
    }
    __syncthreads();

    const int nsteps = kc >> 2;
#pragma unroll 4
    for (int s = 0; s < nsteps; ++s) {
      const int kk = kchunk + 4 * s;
      // A fragment: 16x4, lane lr holds row m0+lr; K = 2*lh + {0,1}
      v2f a;
      a.x = arow[kk + 0];
      a.y = arow[kk + 1];
      // B fragments: rows = (4s + 2*lh) + {0,1} -> pair (2s + lh); one b64 each
      const float* bp = Bs + (2 * s + lh) * PSTRIDE + 2 * lr;
      v2f bf[NACC];
#pragma unroll
      for (int nj = 0; nj < NACC; ++nj) {
        bf[nj] = *(const v2f*)(bp + nj * 32);
      }
#pragma unroll
      for (int nj = 0; nj < NACC; ++nj) {
        acc[nj] = __builtin_amdgcn_wmma_f32_16x16x4_f32(
            false, a, false, bf[nj], (short)0, acc[nj], false, false);
      }
    }
    __syncthreads();
  }

  // Store: C/D 16x16 layout — vgpr r: row = m0 + r + 8*lh, col = nbase + nj*16 + lr
  float* cp = C + (size_t)blockIdx.z * 2048 * Ncols;
#pragma unroll
  for (int nj = 0; nj < NACC; ++nj) {
    if (nbase + nj * 16 < Ncols) {  // wave-uniform guard (Ncols multiple of 16)
      const int col = nbase + nj * 16 + lr;
#pragma unroll
      for (int r = 0; r < 8; ++r) {
        const int row = m0 + r + 8 * lh;
        cp[(size_t)row * Ncols + col] = acc[nj][r];
      }
    }
  }
}

// ---------------------------------------------------------------------------
// Reduce split-K partials, add h @ V, apply leaky-ReLU (leak = 0.2).
// ---------------------------------------------------------------------------
__global__ __launch_bounds__(256) void reduce_bias_lrelu_kernel(
    const float* __restrict__ part, int S,
    const float* __restrict__ h, int cin,
    const float* __restrict__ V,
    float* __restrict__ out, int cout, int total) {
  const int idx = blockIdx.x * 256 + threadIdx.x;
  if (idx >= total) return;
  const int m = idx / cout;
  const int j = idx - m * cout;
  float acc = 0.0f;
  for (int s = 0; s < S; ++s) acc += part[(size_t)s * total + idx];
  const float* hr = h + (size_t)m * cin;
  for (int i = 0; i < cin; ++i) acc = fmaf(hr[i], V[(size_t)i * cout + j], acc);
  out[idx] = fmaxf(acc, 0.2f * acc);
}

// ---------------------------------------------------------------------------
extern "C" void kernel_launch(void* const* d_in, const int* in_sizes, int n_in,
                              void* d_out, int out_size, void* d_ws, size_t ws_size,
                              hipStream_t stream) {
  (void)in_sizes; (void)n_in; (void)out_size; (void)ws_size;
  const float* x   = (const float*)d_in[0];
  const float* adj = (const float*)d_in[1];
  const float* Wl[3] = {(const float*)d_in[2], (const float*)d_in[4], (const float*)d_in[6]};
  const float* Vl[3] = {(const float*)d_in[3], (const float*)d_in[5], (const float*)d_in[7]};

  const int N = 2048, T = 15, S = 16;
  const int Ktot = N * T;            // 30720
  const int Ksl  = Ktot / S;         // 1920

  float* ws   = (float*)d_ws;
  float* hw   = ws;                        // 2048*960 floats (max)
  float* part = hw + (size_t)N * 960;      // S*2048*64 floats (max)
  float* h1   = part + (size_t)S * N * 64; // 2048*64
  float* h2   = h1 + (size_t)N * 64;       // 2048*64

  const int cins[3]  = {32, 64, 64};
  const int couts[3] = {64, 64, 32};
  const float* hin = x;
  float* houts[3] = {h1, h2, (float*)d_out};

  for (int L = 0; L < 3; ++L) {
    const int cin = cins[L], cout = couts[L];
    const int Nhw = T * cout;  // hw columns: 960 / 960 / 480

    // Step A: hw[2048 x Nhw] = h[2048 x cin] * W2[cin x Nhw]  (W permuted on the fly)
    dim3 gA(N / 128, (Nhw + 63) / 64, 1);
    wmma_gemm_kernel<true, 4><<<gA, 256, 0, stream>>>(
        hin, cin, cin, cin, Wl[L], 0, T, cout, hw, Nhw);

    // Step B: part[s][2048 x cout] = adj_slice * hw_slice   (split-K over S slices)
    dim3 gB(N / 128, 1, S);
    if (cout == 64) {
      wmma_gemm_kernel<false, 4><<<gB, 256, 0, stream>>>(
          adj, Ktot, Ktot, Ksl, hw, cout, T, cout, part, cout);
    } else {
      wmma_gemm_kernel<false, 2><<<gB, 256, 0, stream>>>(
          adj, Ktot, Ktot, Ksl, hw, cout, T, cout, part, cout);
    }

    // Step C: reduce partials + h@V + leaky ReLU
    const int total = N * cout;
    reduce_bias_lrelu_kernel<<<(total + 255) / 256, 256, 0, stream>>>(
        part, S, hin, cin, Vl[L], houts[L], cout, total);

    hin = houts[L];
  }
}